// FixedAttention_4131758539224
// MI455X (gfx1250) — compile-verified
//
#include <hip/hip_runtime.h>

// MI455X (gfx1250) fused causal attention, fp32 throughout.
// Memory-bound: 512MB `weights` streamed once (HBM floor ~22us @ 23.3TB/s);
// all 16MB tensors (emb, v, values, weight mats) are L2-resident (192MB L2).
// Matrix ops: V_WMMA_F32_16X16X4_F32 (wave32).
// All global->LDS staging uses GLOBAL_LOAD_ASYNC_TO_LDS_B128 (ASYNCcnt) with
// per-lane LDS destinations, preserving bank-conflict-free padded layouts.

typedef float v2f __attribute__((ext_vector_type(2)));
typedef float v8f __attribute__((ext_vector_type(8)));

#define WMMA_F32(a, b, c) \
  __builtin_amdgcn_wmma_f32_16x16x4_f32(false, (a), false, (b), (short)0, (c), false, false)

constexpr int Bc = 2, Lc = 2048, Ec = 1024, Hc = 16, Ac = 64;

// Generic pointers to LDS carry the DS byte offset in their low 32 bits
// (ISA 10.2 aperture layout).
__device__ __forceinline__ uint32_t lds_addr_of(const void* p) {
  return (uint32_t)(uintptr_t)p;
}

__device__ __forceinline__ void async_copy_b128(const void* lds_dst, const void* gsrc) {
  asm volatile("global_load_async_to_lds_b128 %0, %1, off"
               :: "v"(lds_addr_of(lds_dst)), "v"(gsrc) : "memory");
}

// ---------------------------------------------------------------------------
// C[M][N] = X[M][K] * W[N][K]^T, fp32. 64x64 workgroup tile, K-step 32,
// double-buffered async-to-LDS tile DMA overlapped with WMMA compute.
// LDS row stride 36 -> conflict-free b64 fragment reads, float4 aligned.
// 8 waves, each wave owns two 16x16 C tiles.
// ---------------------------------------------------------------------------
__global__ __launch_bounds__(256)
void gemm_nt_kernel(const float* __restrict__ X, const float* __restrict__ W,
                    float* __restrict__ C, int M, int N, int K) {
  __shared__ float Xs[2][64 * 36];
  __shared__ float Ws[2][64 * 36];
  const int t    = threadIdx.x;
  const int wave = t >> 5, lane = t & 31;
  const int rl   = lane & 15, hi = lane >> 4;
  const int mblk  = wave & 3;   // 16-row slice of the 64-row tile
  const int npair = wave >> 2;  // 32-col slice (two 16-col WMMA tiles)
  const int m0 = blockIdx.x * 64;
  const int n0 = blockIdx.y * 64;

  // 64 rows x 32 cols per operand per K-step: 512 float4s, 2 per thread
  const int r_ld = t >> 3;        // 0..31 (+32 on second issue)
  const int c_ld = (t & 7) << 2;  // 0..28
  auto issue_tile = [&](int kb, int buf) {
#pragma unroll
    for (int j = 0; j < 2; ++j) {
      const int r = r_ld + j * 32;
      async_copy_b128(&Xs[buf][r * 36 + c_ld], X + (size_t)(m0 + r) * K + kb + c_ld);
      async_copy_b128(&Ws[buf][r * 36 + c_ld], W + (size_t)(n0 + r) * K + kb + c_ld);
    }
  };

  v8f c0 = {};
  v8f c1 = {};

  issue_tile(0, 0);
  asm volatile("s_wait_asynccnt 0x0" ::: "memory");
  __syncthreads();

  int buf = 0;
  for (int kb = 0; kb < K; kb += 32) {
    const bool more = (kb + 32) < K;
    if (more) issue_tile(kb + 32, buf ^ 1);  // DMA next tile during compute

    const float* xs = &Xs[buf][0];
    const float* ws = &Ws[buf][0];
#pragma unroll
    for (int kk = 0; kk < 32; kk += 4) {
      const int kc = kk + 2 * hi;  // A/B frag: lo lanes K{0,1}, hi lanes K{2,3}
      v2f a, b0, b1;
      a.x  = xs[(mblk * 16 + rl) * 36 + kc];
      a.y  = xs[(mblk * 16 + rl) * 36 + kc + 1];
      b0.x = ws[(npair * 32 + rl) * 36 + kc];
      b0.y = ws[(npair * 32 + rl) * 36 + kc + 1];
      b1.x = ws[(npair * 32 + 16 + rl) * 36 + kc];
      b1.y = ws[(npair * 32 + 16 + rl) * 36 + kc + 1];
      c0 = WMMA_F32(a, b0, c0);
      c1 = WMMA_F32(a, b1, c1);
    }
    if (more) {
      asm volatile("s_wait_asynccnt 0x0" ::: "memory");
      __syncthreads();
    }
    buf ^= 1;
  }

  // C layout: VGPR r -> M = r + 8*hi, N = lane&15
  const int row_base = m0 + mblk * 16 + 8 * hi;
  const int col0     = n0 + npair * 32 + rl;
#pragma unroll
  for (int rr = 0; rr < 8; ++rr) {
    C[(size_t)(row_base + rr) * N + col0]      = c0[rr];
    C[(size_t)(row_base + rr) * N + col0 + 16] = c1[rr];
  }
}

// ---------------------------------------------------------------------------
// Fused causal softmax + P@V. One workgroup per (b, h, 16-query tile).
// Full 16x2048 score row-block lives in LDS (stride 2052 => conflict-free
// b64 A-fragment reads) -- enabled by the 320KB WGP LDS. The 128KB score
// DMA is issued in 4 column chunks and pipelined against the softmax
// statistics via ordered ASYNCcnt partial waits + online max/sum merge.
// 8 waves split K=2048 (256 each), accumulate 16x64 partials with f32 WMMA,
// combine via LDS float atomics (ds_add_f32), normalize by 1/rowsum.
// ---------------------------------------------------------------------------
__global__ __launch_bounds__(256)
void attn_kernel(const float* __restrict__ Wt, const float* __restrict__ Vb,
                 float* __restrict__ Out) {
  __shared__ float wt[16 * 2052];   // 128.2 KB score tile (fp32)
  __shared__ float pacc[16 * 64];   // 4 KB cross-wave accumulator
  __shared__ float rowm[16];
  __shared__ float rinv[16];

  const int t   = threadIdx.x;
  const int gid = blockIdx.x;
  const int qt  = gid & 127;         // L/16 tiles
  const int h   = (gid >> 7) & 15;
  const int b   = gid >> 11;
  const int q0  = qt * 16;

  // ---- phase 1: issue all score DMA in column-chunk order ----------------
  // 4 chunks x (16 rows x 512 cols); 8 async b128 per thread per chunk.
  const float* wrow = Wt + (((size_t)(b * Hc + h) * Lc) + q0) * Lc;
  for (int c = 0; c < 4; ++c) {
    for (int i = t; i < 16 * 128; i += 256) {
      const int r = i >> 7;
      const int col = c * 512 + ((i & 127) << 2);
      async_copy_b128(wt + r * 2052 + col, wrow + (size_t)r * Lc + col);
    }
  }
  for (int i = t; i < 16 * 64; i += 256) pacc[i] = 0.0f;

  // ---- phase 2: chunked online row max/sum, overlapped with the DMA ------
  {
    const int row = t >> 4, j = t & 15, q = q0 + row;
    float m = -3.0e38f, s = 0.0f;
#pragma unroll
    for (int c = 0; c < 4; ++c) {
      // async loads retire in order: chunk c resident once <= (3-c)*8 remain
      asm volatile("s_wait_asynccnt %0" :: "i"((3 - c) * 8) : "memory");
      __syncthreads();
      const int k0 = c * 512, kend = k0 + 512;
      float mc = -3.0e38f;
      for (int k = k0 + j; k < kend; k += 16)
        if (k <= q) mc = fmaxf(mc, wt[row * 2052 + k]);
      for (int off = 8; off; off >>= 1) mc = fmaxf(mc, __shfl_xor(mc, off, 16));
      float sc = 0.0f;
      for (int k = k0 + j; k < kend; k += 16)
        if (k <= q) sc += __expf(wt[row * 2052 + k] - mc);
      for (int off = 8; off; off >>= 1) sc += __shfl_xor(sc, off, 16);
      const float M = fmaxf(m, mc);
      s = s * __expf(m - M) + sc * __expf(mc - M);
      m = M;
    }
    if (j == 0) { rowm[row] = m; rinv[row] = 1.0f / s; }
  }
  __syncthreads();

  // ---- phase 3: P @ V with WMMA f32 16x16x4 ------------------------------
  const int wave = t >> 5, lane = t & 31;
  const int n = lane & 15, hi = lane >> 4;  // n = P row (M) AND V col (N)
  const float mrow = rowm[n];
  const int q = q0 + n;
  v8f acc0 = {}, acc1 = {}, acc2 = {}, acc3 = {};
  const float* vb = Vb + (size_t)b * Lc * Ec + h * Ac;  // + k*Ec + a

  for (int kk = 0; kk < 256; kk += 4) {
    const int k = wave * 256 + kk + 2 * hi;
    const float w0 = wt[n * 2052 + k];
    const float w1 = wt[n * 2052 + k + 1];
    v2f a;
    a.x = (k     <= q) ? __expf(w0 - mrow) : 0.0f;  // unnormalized prob
    a.y = (k + 1 <= q) ? __expf(w1 - mrow) : 0.0f;
    const float* vk = vb + (size_t)k * Ec;
    v2f b0, b1, b2, b3;
    b0.x = vk[ 0 + n]; b0.y = vk[Ec +  0 + n];
    b1.x = vk[16 + n]; b1.y = vk[Ec + 16 + n];
    b2.x = vk[32 + n]; b2.y = vk[Ec + 32 + n];
    b3.x = vk[48 + n]; b3.y = vk[Ec + 48 + n];
    acc0 = WMMA_F32(a, b0, acc0);
    acc1 = WMMA_F32(a, b1, acc1);
    acc2 = WMMA_F32(a, b2, acc2);
    acc3 = WMMA_F32(a, b3, acc3);
  }

  // combine the 8 per-wave partials via LDS float atomics (ds_add_f32)
#pragma unroll
  for (int rr = 0; rr < 8; ++rr) {
    const int prow = rr + 8 * hi;
    atomicAdd(&pacc[prow * 64 +  0 + n], acc0[rr]);
    atomicAdd(&pacc[prow * 64 + 16 + n], acc1[rr]);
    atomicAdd(&pacc[prow * 64 + 32 + n], acc2[rr]);
    atomicAdd(&pacc[prow * 64 + 48 + n], acc3[rr]);
  }
  __syncthreads();

  // ---- phase 4: normalize + store [b][q][h*64+a] -------------------------
  for (int i = t; i < 16 * 64; i += 256) {
    const int row = i >> 6, a = i & 63;
    Out[(size_t)(b * Lc + q0 + row) * Ec + h * Ac + a] = pacc[i] * rinv[row];
  }
}

// ---------------------------------------------------------------------------
extern "C" void kernel_launch(void* const* d_in, const int* in_sizes, int n_in,
                              void* d_out, int out_size, void* d_ws, size_t ws_size,
                              hipStream_t stream) {
  (void)in_sizes; (void)n_in; (void)out_size; (void)ws_size;
  const float* emb  = (const float*)d_in[0];  // [B,L,E]
  const float* wts  = (const float*)d_in[1];  // [B,H,L,L]
  // d_in[2] = causal_mask: implicit (k <= q), not read
  const float* V_w  = (const float*)d_in[3];  // [E,E]
  const float* outw = (const float*)d_in[4];  // [E,E]

  float* vbuf = (float*)d_ws;                 // v = emb @ V_w^T : [B*L][E] (16 MB)
  float* vals = vbuf + (size_t)Bc * Lc * Ec;  // attention out    : [B*L][E] (16 MB)

  dim3 g(Bc * Lc / 64, Ec / 64);
  gemm_nt_kernel<<<g, 256, 0, stream>>>(emb, V_w, vbuf, Bc * Lc, Ec, Ec);
  attn_kernel<<<Bc * Hc * (Lc / 16), 256, 0, stream>>>(wts, vbuf, vals);
  gemm_nt_kernel<<<g, 256, 0, stream>>>(vals, outw, (float*)d_out, Bc * Lc, Ec, Ec);
}